// VectorQuantizer_13649406067019
// MI455X (gfx1250) — compile-verified
//
#include <hip/hip_runtime.h>
#include <hip/hip_bf16.h>
#include <math.h>

typedef __attribute__((ext_vector_type(2))) float v2f;
typedef __attribute__((ext_vector_type(8))) float v8f;

#define B_DIM 32
#define C_DIM 64
#define HW    1024            // H*W = 32*32
#define K_DIM 1024
#define N_TOT 32768           // B*HW

// d_out layout (floats): [0,2097152) quantized, [2097152] loss,
//                        [2097153, 2097153+32768) indices (as f32), [2129921] perplexity
#define OUT_Q_ELEMS (N_TOT * C_DIM)
#define OUT_LOSS    OUT_Q_ELEMS
#define OUT_IDX     (OUT_Q_ELEMS + 1)
#define OUT_PERP    (OUT_Q_ELEMS + 1 + N_TOT)

// d_ws layout (floats): [0,1024) counts, [1024] sse, [1025,2049) enorm
#define WS_COUNTS 0
#define WS_SSE    1024
#define WS_ENORM  1025

// ---- kernel 1: e-norms + zero counts/sse -----------------------------------
__global__ __launch_bounds__(256) void vq_prep(const float* __restrict__ embed,
                                               float* __restrict__ ws) {
  int j = blockIdx.x * 256 + threadIdx.x;
  if (j < K_DIM) {
    float s = 0.f;
#pragma unroll
    for (int c = 0; c < C_DIM; ++c) {
      float e = embed[(size_t)c * K_DIM + j];
      s += e * e;
    }
    ws[WS_ENORM + j] = s;
    ws[WS_COUNTS + j] = 0.f;
    if (j == 0) ws[WS_SSE] = 0.f;
  }
}

// ---- kernel 2: fused GEMM(WMMA f32) + argmin + gather + SSE + counts -------
__global__ __launch_bounds__(256) void vq_main(const float* __restrict__ in,    // (B,C,H,W)
                                               const float* __restrict__ embed, // (C,K)
                                               float* __restrict__ out,
                                               float* __restrict__ ws) {
  const float* enorm  = ws + WS_ENORM;
  float*       counts = ws + WS_COUNTS;
  float*       sse    = ws + WS_SSE;

  const int tid  = threadIdx.x;
  const int lane = tid & 31;
  const int wave = tid >> 5;        // 0..7
  const int half = lane >> 4;       // 0 or 1 (K-split of f32 WMMA A/B layout)
  const int l16  = lane & 15;

  const int rowBase = blockIdx.x * 16;      // 16 rows (flat view) per block
  const int b       = rowBase >> 10;        // rowBase / HW; whole block shares b
  const int hwBase  = rowBase & (HW - 1);

  // ---- A tile: flat[row][k] = in[b*C*HW + k*HW + hw]; ISA 16x4 f32 A layout:
  //      half 0 holds K = {4kt, 4kt+1}, half 1 holds K = {4kt+2, 4kt+3}, M = l16
  const float* inRow = in + (size_t)b * C_DIM * HW + (hwBase + l16);
  v2f a[16];
#pragma unroll
  for (int kt = 0; kt < 16; ++kt) {
    const int k0 = 4 * kt + (half ? 2 : 0);
    a[kt].x = inRow[(size_t)k0 * HW];
    a[kt].y = inRow[(size_t)(k0 + 1) * HW];
  }

  float bestVal[8];
  int   bestIdx[8];
#pragma unroll
  for (int r = 0; r < 8; ++r) { bestVal[r] = 3.4e38f; bestIdx[r] = 0; }

  // wave w sweeps column tiles [8w, 8w+8) -> cols [128w, 128w+128)
  for (int i = 0; i < 8; ++i) {
    const int colBase = (wave * 8 + i) * 16;
    const int myCol   = colBase + l16;

    v2f bm[16];
#pragma unroll
    for (int kt = 0; kt < 16; ++kt) {
      const int k0 = 4 * kt + (half ? 2 : 0);
      bm[kt].x = embed[(size_t)k0 * K_DIM + myCol];
      bm[kt].y = embed[(size_t)(k0 + 1) * K_DIM + myCol];
    }

    v8f acc = {};
#pragma unroll
    for (int kt = 0; kt < 16; ++kt) {
      // D = A x B + C, f32 16x16x4; accumulator chained through SRC2
      acc = __builtin_amdgcn_wmma_f32_16x16x4_f32(
          /*neg_a=*/false, a[kt], /*neg_b=*/false, bm[kt],
          /*c_mod=*/(short)0, acc, /*reuse_a=*/false, /*reuse_b=*/false);
    }

    // dist = ||e||^2 - 2*x.e   (||x||^2 dropped: constant per row)
    const float en = enorm[myCol];
#pragma unroll
    for (int r = 0; r < 8; ++r) {
      const float d = en - 2.0f * acc[r];
      if (d < bestVal[r]) { bestVal[r] = d; bestIdx[r] = myCol; }
    }
  }

  // intra-wave argmin over 16 lanes of each half (vgpr r -> row r / r+8)
#pragma unroll
  for (int r = 0; r < 8; ++r) {
#pragma unroll
    for (int off = 1; off < 16; off <<= 1) {
      const float ov = __shfl_xor(bestVal[r], off, 32);
      const int   oi = __shfl_xor(bestIdx[r], off, 32);
      if (ov < bestVal[r] || (ov == bestVal[r] && oi < bestIdx[r])) {
        bestVal[r] = ov; bestIdx[r] = oi;
      }
    }
  }

  __shared__ float sVal[8][16];
  __shared__ int   sIdx[8][16];
  __shared__ int   sFin[16];
  if (l16 == 0) {
#pragma unroll
    for (int r = 0; r < 8; ++r) {
      sVal[wave][half * 8 + r] = bestVal[r];
      sIdx[wave][half * 8 + r] = bestIdx[r];
    }
  }
  __syncthreads();

  if (tid < 16) {   // final cross-wave argmin for row `tid`; tie -> lowest idx
    float bv = sVal[0][tid];
    int   bi = sIdx[0][tid];
#pragma unroll
    for (int w = 1; w < 8; ++w) {
      const float v  = sVal[w][tid];
      const int   ix = sIdx[w][tid];
      if (v < bv || (v == bv && ix < bi)) { bv = v; bi = ix; }
    }
    sFin[tid] = bi;
    out[OUT_IDX + rowBase + tid] = (float)bi;
    atomicAdd(&counts[bi], 1.0f);
  }
  __syncthreads();

  // gather embed.T[idx], scatter to (B,C,H,W), accumulate SSE
  const int row = tid & 15;
  const int c0  = tid >> 4;              // 0..15
  const int hw  = hwBase + row;
  const int qi  = sFin[row];
  float part = 0.f;
#pragma unroll
  for (int j = 0; j < 4; ++j) {
    const int    c   = c0 + 16 * j;
    const size_t off = (size_t)b * C_DIM * HW + (size_t)c * HW + hw;
    const float  q   = embed[(size_t)c * K_DIM + qi];
    const float  x   = in[off];
    const float  df  = q - x;
    part += df * df;
    out[off] = q;                        // quantized_st == quantized numerically
  }
#pragma unroll
  for (int off = 1; off < 32; off <<= 1) part += __shfl_xor(part, off, 32);
  if (lane == 0) atomicAdd(sse, part);
}

// ---- kernel 3: loss + perplexity -------------------------------------------
__global__ __launch_bounds__(1024) void vq_final(const float* __restrict__ ws,
                                                 float* __restrict__ out) {
  __shared__ float red[1024];
  const int t = threadIdx.x;
  const float p = ws[WS_COUNTS + t] * (1.0f / (float)N_TOT);
  red[t] = p * logf(p + 1e-10f);
  __syncthreads();
  for (int s = 512; s > 0; s >>= 1) {
    if (t < s) red[t] += red[t + s];
    __syncthreads();
  }
  if (t == 0) {
    out[OUT_LOSS] = 0.25f * ws[WS_SSE] / (float)((size_t)N_TOT * C_DIM);
    out[OUT_PERP] = expf(-red[0]);
  }
}

extern "C" void kernel_launch(void* const* d_in, const int* in_sizes, int n_in,
                              void* d_out, int out_size, void* d_ws, size_t ws_size,
                              hipStream_t stream) {
  (void)in_sizes; (void)n_in; (void)out_size; (void)ws_size;
  const float* in    = (const float*)d_in[0];   // (32,64,32,32) f32
  const float* embed = (const float*)d_in[1];   // (64,1024) f32
  float* out = (float*)d_out;
  float* ws  = (float*)d_ws;

  vq_prep<<<K_DIM / 256, 256, 0, stream>>>(embed, ws);
  vq_main<<<N_TOT / 16, 256, 0, stream>>>(in, embed, out, ws);
  vq_final<<<1, 1024, 0, stream>>>(ws, out);
}